// GAT_55671366091333
// MI455X (gfx1250) — compile-verified
//
#include <hip/hip_runtime.h>
#include <math.h>

typedef float v2f __attribute__((ext_vector_type(2)));
typedef float v8f __attribute__((ext_vector_type(8)));

#define NN      50000     // nodes
#define NE      800000    // explicit edges
#define ETOT    (NE + NN) // edges + self loops
#define NEG_SLOPE 0.2f
#define GAT_EPS 1e-16f

// ---------------------------------------------------------------------------
// FP32 WMMA GEMM: C[M,N] = A[M,K] * B[K,N], row-major. N, K compile-time.
// One wave -> 32x32 tile: 2 A fragments x 2 B fragments -> 4 f32 accumulators.
// 32-bit indices from uniform base pointers => saddr+voffset addressing with
// immediate offsets; one 32-bit index bump per k-step.
// blockDim.x = 32*(N/32); gridDim.x = ceil(M/32). M%16==0, N%32==0, K%4==0.
// ---------------------------------------------------------------------------
template<int N, int K>
__global__ void gemm_wmma_f32(const float* __restrict__ A,
                              const float* __restrict__ B,
                              float* __restrict__ C, int M)
{
    const int lane  = threadIdx.x & 31;
    const int nt    = threadIdx.x >> 5;       // n-tile within block
    const int m0    = blockIdx.x << 5;        // 32 rows per wave tile
    const int n0    = nt << 5;                // 32 cols per wave tile

    const int l15   = lane & 15;
    const int khalf = (lane >> 4) << 1;       // 0 or 2 (A/B fragment K pair)
    const int ncol  = n0 + l15;

    const int mrow0 = m0 + l15;
    int mrow1 = m0 + 16 + l15;
    if (mrow1 >= M) mrow1 = M - 1;            // clamp; results discarded

    unsigned aOff0 = (unsigned)mrow0 * K + khalf;
    unsigned aOff1 = (unsigned)mrow1 * K + khalf;
    unsigned bOff  = (unsigned)khalf * N + ncol;

    v8f acc00 = {}, acc01 = {}, acc10 = {}, acc11 = {};

    #pragma unroll 4
    for (int k = 0; k < K; k += 4) {
        v2f a0 = *(const v2f*)(A + aOff0);     // b64: K pair for rows 0..15
        v2f a1 = *(const v2f*)(A + aOff1);     // b64: K pair for rows 16..31
        v2f b0, b1;                            // 4x16 fragments, 2 n-subtiles
        b0.x = B[bOff];
        b0.y = B[bOff + N];
        b1.x = B[bOff + 16];
        b1.y = B[bOff + N + 16];
        aOff0 += 4; aOff1 += 4; bOff += 4 * N;

        acc00 = __builtin_amdgcn_wmma_f32_16x16x4_f32(false, a0, false, b0,
                                                      (short)0, acc00, false, false);
        acc01 = __builtin_amdgcn_wmma_f32_16x16x4_f32(false, a0, false, b1,
                                                      (short)0, acc01, false, false);
        acc10 = __builtin_amdgcn_wmma_f32_16x16x4_f32(false, a1, false, b0,
                                                      (short)0, acc10, false, false);
        acc11 = __builtin_amdgcn_wmma_f32_16x16x4_f32(false, a1, false, b1,
                                                      (short)0, acc11, false, false);
    }

    // C/D layout: VGPR r -> M = mbase + r + 8*(lane>=16), N = n0 + (lane&15)
    const int srow0 = m0 + ((lane >> 4) << 3);
    const int scol  = n0 + l15;
    #pragma unroll
    for (int r = 0; r < 8; ++r) {
        C[(size_t)(srow0 + r) * N + scol]      = acc00[r];
        C[(size_t)(srow0 + r) * N + scol + 16] = acc01[r];
    }
    if (m0 + 16 < M) {                        // second m-subtile fully valid?
        const int srow1 = srow0 + 16;
        #pragma unroll
        for (int r = 0; r < 8; ++r) {
            C[(size_t)(srow1 + r) * N + scol]      = acc10[r];
            C[(size_t)(srow1 + r) * N + scol + 16] = acc11[r];
        }
    }
}

// ---------------------------------------------------------------------------
// Per-node attention scores: asrc[i] = h[i].att_src, adst[i] = h[i].att_dst
// One wave per node.
// ---------------------------------------------------------------------------
template<int F>
__global__ void scores_kernel(const float* __restrict__ h,
                              const float* __restrict__ att_src,
                              const float* __restrict__ att_dst,
                              float* __restrict__ asrc,
                              float* __restrict__ adst)
{
    const int wid  = (blockIdx.x * blockDim.x + threadIdx.x) >> 5;
    const int lane = threadIdx.x & 31;
    if (wid >= NN) return;

    const float* row = h + (size_t)wid * F;
    float ss = 0.f, sd = 0.f;
    #pragma unroll
    for (int i = 0; i < F / 32; ++i) {
        float v = row[lane + i * 32];
        ss += v * att_src[lane + i * 32];
        sd += v * att_dst[lane + i * 32];
    }
    #pragma unroll
    for (int off = 16; off > 0; off >>= 1) {
        ss += __shfl_down(ss, off, 32);
        sd += __shfl_down(sd, off, 32);
    }
    if (lane == 0) { asrc[wid] = ss; adst[wid] = sd; }
}

// ---------------------------------------------------------------------------
// Init: m = -inf, s = 0 per node
// ---------------------------------------------------------------------------
__global__ void init_ms_kernel(float* __restrict__ m, float* __restrict__ s)
{
    int i = blockIdx.x * blockDim.x + threadIdx.x;
    if (i < NN) { m[i] = __int_as_float(0xFF800000); s[i] = 0.f; }
}

__global__ void zero_kernel(float* __restrict__ p, long long n)
{
    long long i = (long long)blockIdx.x * blockDim.x + threadIdx.x;
    long long stride = (long long)gridDim.x * blockDim.x;
    for (; i < n; i += stride) p[i] = 0.f;
}

__device__ __forceinline__ void atomic_max_f32(float* addr, float v)
{
    if (v >= 0.f) atomicMax((int*)addr, __float_as_int(v));
    else          atomicMin((unsigned int*)addr, (unsigned int)__float_as_int(v));
}

// ---------------------------------------------------------------------------
// Pass 1: edge logits (leaky relu) + segment max over dst
// ---------------------------------------------------------------------------
__global__ void edge_max_kernel(const long long* __restrict__ src,
                                const long long* __restrict__ dst,
                                const float* __restrict__ asrc,
                                const float* __restrict__ adst,
                                float* __restrict__ elog,
                                float* __restrict__ m)
{
    int e = blockIdx.x * blockDim.x + threadIdx.x;
    if (e >= ETOT) return;
    int s, d;
    if (e < NE) { s = (int)src[e]; d = (int)dst[e]; }
    else        { s = d = e - NE; }
    float v = asrc[s] + adst[d];
    v = v > 0.f ? v : NEG_SLOPE * v;
    elog[e] = v;
    atomic_max_f32(m + d, v);
}

// ---------------------------------------------------------------------------
// Pass 2: ex = exp(logit - max[dst]); segment sum over dst; elog <- ex
// ---------------------------------------------------------------------------
__global__ void edge_sum_kernel(const long long* __restrict__ dst,
                                float* __restrict__ elog,
                                const float* __restrict__ m,
                                float* __restrict__ s)
{
    int e = blockIdx.x * blockDim.x + threadIdx.x;
    if (e >= ETOT) return;
    int d = (e < NE) ? (int)dst[e] : (e - NE);
    float ex = __expf(elog[e] - m[d]);
    elog[e] = ex;
    atomicAdd(s + d, ex);
}

// ---------------------------------------------------------------------------
// Pass 3: out[dst] += alpha * h[src]; one wave per edge, float4 per lane chunk
// ---------------------------------------------------------------------------
template<int F>
__global__ void aggregate_kernel(const long long* __restrict__ src,
                                 const long long* __restrict__ dst,
                                 const float* __restrict__ elog,
                                 const float* __restrict__ ssum,
                                 const float* __restrict__ h,
                                 float* __restrict__ out)
{
    const int wid  = (blockIdx.x * blockDim.x + threadIdx.x) >> 5;
    const int lane = threadIdx.x & 31;
    if (wid >= ETOT) return;
    int s, d;
    if (wid < NE) { s = (int)src[wid]; d = (int)dst[wid]; }
    else          { s = d = wid - NE; }
    const float alpha = elog[wid] / (ssum[d] + GAT_EPS);

    const float4* hp = (const float4*)(h + (size_t)s * F);
    float* op = out + (size_t)d * F;
    #pragma unroll
    for (int i = 0; i < F / 128; ++i) {
        float4 v = hp[lane + i * 32];
        int base = (lane + i * 32) << 2;
        atomicAdd(op + base + 0, alpha * v.x);
        atomicAdd(op + base + 1, alpha * v.y);
        atomicAdd(op + base + 2, alpha * v.z);
        atomicAdd(op + base + 3, alpha * v.w);
    }
}

// ---------------------------------------------------------------------------
// x = relu(x + bias[f])  (F must be a power of two)
// ---------------------------------------------------------------------------
template<int F>
__global__ void bias_relu_kernel(float* __restrict__ x,
                                 const float* __restrict__ b,
                                 long long total)
{
    long long i = (long long)blockIdx.x * blockDim.x + threadIdx.x;
    if (i >= total) return;
    float v = x[i] + b[i & (F - 1)];
    x[i] = v > 0.f ? v : 0.f;
}

// ---------------------------------------------------------------------------
// Final: h = agg + bias (in place, first half of d_out);
//        softmax(h, axis=1) into second half. One wave per node, F=128.
// ---------------------------------------------------------------------------
__global__ void bias_softmax_kernel(float* __restrict__ out,
                                    const float* __restrict__ b)
{
    const int wid  = (blockIdx.x * blockDim.x + threadIdx.x) >> 5;
    const int lane = threadIdx.x & 31;
    if (wid >= NN) return;

    float* row = out + (size_t)wid * 128;
    float v[4];
    float mx = __int_as_float(0xFF800000);
    #pragma unroll
    for (int i = 0; i < 4; ++i) {
        v[i] = row[lane * 4 + i] + b[lane * 4 + i];
        mx = fmaxf(mx, v[i]);
    }
    #pragma unroll
    for (int off = 16; off > 0; off >>= 1)
        mx = fmaxf(mx, __shfl_xor(mx, off, 32));

    float ex[4], sum = 0.f;
    #pragma unroll
    for (int i = 0; i < 4; ++i) { ex[i] = __expf(v[i] - mx); sum += ex[i]; }
    #pragma unroll
    for (int off = 16; off > 0; off >>= 1)
        sum += __shfl_xor(sum, off, 32);
    const float inv = 1.f / sum;

    float* srow = out + (size_t)NN * 128 + (size_t)wid * 128;
    #pragma unroll
    for (int i = 0; i < 4; ++i) {
        row[lane * 4 + i]  = v[i];
        srow[lane * 4 + i] = ex[i] * inv;
    }
}

// ---------------------------------------------------------------------------
extern "C" void kernel_launch(void* const* d_in, const int* in_sizes, int n_in,
                              void* d_out, int out_size, void* d_ws, size_t ws_size,
                              hipStream_t stream)
{
    const float*     x     = (const float*)d_in[0];
    const long long* ei    = (const long long*)d_in[1]; // int64 [2, NE]
    const float*     W1    = (const float*)d_in[2];
    const float*     as1   = (const float*)d_in[3];
    const float*     ad1   = (const float*)d_in[4];
    const float*     b1    = (const float*)d_in[5];
    const float*     W2    = (const float*)d_in[6];
    const float*     as2   = (const float*)d_in[7];
    const float*     ad2   = (const float*)d_in[8];
    const float*     b2    = (const float*)d_in[9];

    const long long* src = ei;
    const long long* dst = ei + NE;

    // workspace carve-up (floats)
    float* ws   = (float*)d_ws;
    float* h1   = ws;                       // NN*256
    float* hr   = h1 + (size_t)NN * 256;    // NN*256 (agg1 -> relu out)
    float* h2   = hr + (size_t)NN * 256;    // NN*128
    float* asc  = h2 + (size_t)NN * 128;    // NN
    float* adc  = asc + NN;                 // NN
    float* mbuf = adc + NN;                 // NN
    float* sbuf = mbuf + NN;                // NN
    float* elog = sbuf + NN;                // ETOT

    float* outp = (float*)d_out;            // [NN*128 h] ++ [NN*128 softmax]

    const int mt32 = (NN + 31) / 32;                 // 32-row GEMM tiles
    const int eb   = (ETOT + 255) / 256;             // edge 1-thread blocks
    const int ewb  = (ETOT * 32 + 255) / 256;        // edge 1-wave blocks
    const int nwb  = (NN * 32 + 255) / 256;          // node 1-wave blocks
    const int nb   = (NN + 255) / 256;

    // ---------------- Layer 1 ----------------
    gemm_wmma_f32<256, 512><<<mt32, 256, 0, stream>>>(x, W1, h1, NN);
    scores_kernel<256><<<nwb, 256, 0, stream>>>(h1, as1, ad1, asc, adc);
    init_ms_kernel<<<nb, 256, 0, stream>>>(mbuf, sbuf);
    edge_max_kernel<<<eb, 256, 0, stream>>>(src, dst, asc, adc, elog, mbuf);
    edge_sum_kernel<<<eb, 256, 0, stream>>>(dst, elog, mbuf, sbuf);
    zero_kernel<<<2048, 256, 0, stream>>>(hr, (long long)NN * 256);
    aggregate_kernel<256><<<ewb, 256, 0, stream>>>(src, dst, elog, sbuf, h1, hr);
    bias_relu_kernel<256><<<(int)(((long long)NN * 256 + 255) / 256), 256, 0, stream>>>(
        hr, b1, (long long)NN * 256);

    // ---------------- Layer 2 ----------------
    gemm_wmma_f32<128, 256><<<mt32, 128, 0, stream>>>(hr, W2, h2, NN);
    scores_kernel<128><<<nwb, 256, 0, stream>>>(h2, as2, ad2, asc, adc);
    init_ms_kernel<<<nb, 256, 0, stream>>>(mbuf, sbuf);
    edge_max_kernel<<<eb, 256, 0, stream>>>(src, dst, asc, adc, elog, mbuf);
    edge_sum_kernel<<<eb, 256, 0, stream>>>(dst, elog, mbuf, sbuf);
    zero_kernel<<<2048, 256, 0, stream>>>(outp, (long long)NN * 128);
    aggregate_kernel<128><<<ewb, 256, 0, stream>>>(src, dst, elog, sbuf, h2, outp);
    bias_softmax_kernel<<<nwb, 256, 0, stream>>>(outp, b2);
}